// SiameseLDDTLoss_20796231647443
// MI455X (gfx1250) — compile-verified
//
#include <hip/hip_runtime.h>

// ---------------------------------------------------------------------------
// SiameseLDDTLoss on MI455X (gfx1250, wave32)
//
//   residue: for 8192 alignment pairs, sim = <v1_hat, v2_hat>, MSE vs lddt.
//   Wave-tile = 16 pairs. Gathered rows are DMA'd straight into LDS with
//   GLOBAL_LOAD_ASYNC_TO_LDS_B128 (ASYNCcnt path, no VGPR round-trip), rows
//   padded to 132 floats for bank-conflict-free fragment reads. Dot products
//   = diagonal of V1 * V2^T via V_WMMA_F32_16X16X4_F32 (full f32 precision);
//   row norms run on the VALU in parallel. One atomicAdd per wave into the
//   workspace; a tiny finalize kernel adds the global cosine term and writes
//   (total, residue, global).
// ---------------------------------------------------------------------------

typedef __attribute__((ext_vector_type(2))) float v2f;
typedef __attribute__((ext_vector_type(8))) float v8f;

#define B_    8
#define L_    2048
#define D_    128
#define A_    1024
#define ROWP  132          // padded row length in floats (132 % 64 == 4)
#define TILE  16           // pairs per wave
#define WPB   2            // waves per block
#define NPAIR (B_ * A_)    // 8192
#define EPS_  1e-12f

__global__ void sll_zero_ws(float* ws) {
    if (threadIdx.x == 0) ws[0] = 0.0f;
}

__global__ __launch_bounds__(WPB * 32)
void sll_residue_kernel(const float* __restrict__ e1,
                        const float* __restrict__ e2,
                        const float* __restrict__ lddt,
                        const int*   __restrict__ align,
                        float*       __restrict__ acc) {
    __shared__ float lds[WPB][2][TILE * ROWP];

    const int lane = threadIdx.x & 31;
    const int wave = threadIdx.x >> 5;
    const int tile = blockIdx.x * WPB + wave;
    const int p0   = tile * TILE;          // first pair of this wave-tile
    const int b    = p0 / A_;              // whole tile shares one batch (A%16==0)
    const int a0   = p0 % A_;

    // lanes 0..15 own pair t = lane (16..31 mirror them; keeps EXEC uniform)
    const int t16    = lane & 15;
    const int abase  = (b * A_ + a0 + t16) * 2;
    const int i_idx  = align[abase + 0];
    const int j_idx  = align[abase + 1];
    const float lddt_t = lddt[b * L_ + i_idx];

    float* ldsA = &lds[wave][0][0];
    float* ldsB = &lds[wave][1][0];

    // ---- async DMA: gather 16 v1 rows + 16 v2 rows straight into LDS ------
    // Each lane copies 16B of the current 512B row (coalesced), tracked on
    // ASYNCcnt; all 32 row transfers overlap, then one s_wait_asynccnt 0.
    {
        const uint32_t lA0 = (uint32_t)(size_t)ldsA + (uint32_t)(lane * 16);
        const uint32_t lB0 = (uint32_t)(size_t)ldsB + (uint32_t)(lane * 16);
        #pragma unroll 4
        for (int t = 0; t < TILE; ++t) {
            const int it = __shfl(i_idx, t);
            const int jt = __shfl(j_idx, t);
            const float* pga = e1 + ((size_t)b * L_ + it) * D_ + lane * 4;
            const float* pgb = e2 + ((size_t)b * L_ + jt) * D_ + lane * 4;
            const uint32_t la = lA0 + (uint32_t)(t * (ROWP * 4));
            const uint32_t lb = lB0 + (uint32_t)(t * (ROWP * 4));
            asm volatile("global_load_async_to_lds_b128 %0, %1, off"
                         :: "v"(la), "v"(pga) : "memory");
            asm volatile("global_load_async_to_lds_b128 %0, %1, off"
                         :: "v"(lb), "v"(pgb) : "memory");
        }
        asm volatile("s_wait_asynccnt 0x0" ::: "memory");
    }

    // ---- row norms on the VALU: lane handles row lane%16, half lane/16 ----
    const int  row  = lane & 15;
    const int  half = lane >> 4;
    float na = 0.0f, nb = 0.0f;
    {
        const float* pa = ldsA + row * ROWP + half * 64;
        const float* pb = ldsB + row * ROWP + half * 64;
        #pragma unroll
        for (int q = 0; q < 64; q += 4) {
            const float4 x = *(const float4*)(pa + q);
            const float4 y = *(const float4*)(pb + q);
            na += x.x * x.x + x.y * x.y + x.z * x.z + x.w * x.w;
            nb += y.x * y.x + y.y * y.y + y.z * y.z + y.w * y.w;
        }
        na += __shfl_xor(na, 16);   // full ||v1_row||^2 in every lane of the pair
        nb += __shfl_xor(nb, 16);
    }

    // ---- 16x16 = V1 * V2^T via 32 x V_WMMA_F32_16X16X4_F32 ----
    // A frag (16x4): lane row=lane%16, K = {0,1}+2*(lane>=16)  -> 2 VGPRs
    // B frag (4x16): N = lane%16,      K = {0,1}+2*(lane>=16)  (B = V2^T)
    const float* fa = ldsA + row * ROWP + (half << 1);
    const float* fb = ldsB + row * ROWP + (half << 1);
    v8f c = {};
    #pragma unroll
    for (int k0 = 0; k0 < D_; k0 += 4) {
        const v2f afrag = *(const v2f*)(fa + k0);
        const v2f bfrag = *(const v2f*)(fb + k0);
        c = __builtin_amdgcn_wmma_f32_16x16x4_f32(
                false, afrag, false, bfrag, (short)0, c, false, false);
    }

    // ---- extract diagonal: (m,m) at (vgpr m, lane m) for m<8,
    //                               (vgpr m-8, lane m+16) for m>=8.
    // Active lanes: 0..7 (m=lane) and 24..31 (m=lane-16); both use vgpr lane&7.
    const int r7 = lane & 7;
    float diag = 0.0f;
    #pragma unroll
    for (int r = 0; r < 8; ++r) {
        diag = (r7 == r) ? c[r] : diag;
    }
    const bool active = (lane < 8) || (lane >= 24);   // pair m = lane & 15

    const float lddt_m = __shfl(lddt_t, row);          // pair m's lddt (held by lane m)
    const float inv = 1.0f / (fmaxf(__builtin_sqrtf(na), EPS_) *
                              fmaxf(__builtin_sqrtf(nb), EPS_));
    const float err  = diag * inv - lddt_m;
    float part = active ? err * err : 0.0f;

    // ---- wave tree-reduce, one atomic per wave ----
    #pragma unroll
    for (int off = 16; off > 0; off >>= 1) part += __shfl_xor(part, off);
    if (lane == 0) atomicAdd(acc, part);
}

__global__ __launch_bounds__(256)
void sll_finalize_kernel(const float* __restrict__ g1,
                         const float* __restrict__ g2,
                         const float* __restrict__ acc,
                         float*       __restrict__ out) {
    __shared__ float gl[B_];
    const int lane = threadIdx.x & 31;
    const int wave = threadIdx.x >> 5;   // wave == batch b (8 waves, B_=8)

    const float4 x = *(const float4*)(g1 + wave * D_ + lane * 4);
    const float4 y = *(const float4*)(g2 + wave * D_ + lane * 4);
    float dot = x.x * y.x + x.y * y.y + x.z * y.z + x.w * y.w;
    float n1  = x.x * x.x + x.y * x.y + x.z * x.z + x.w * x.w;
    float n2  = y.x * y.x + y.y * y.y + y.z * y.z + y.w * y.w;
    #pragma unroll
    for (int off = 16; off > 0; off >>= 1) {
        dot += __shfl_xor(dot, off);
        n1  += __shfl_xor(n1,  off);
        n2  += __shfl_xor(n2,  off);
    }
    if (lane == 0) {
        const float cosv = dot / (fmaxf(__builtin_sqrtf(n1), EPS_) *
                                  fmaxf(__builtin_sqrtf(n2), EPS_));
        const float e = cosv - 0.8f;
        gl[wave] = e * e;
    }
    __syncthreads();
    if (threadIdx.x == 0) {
        float g = 0.0f;
        #pragma unroll
        for (int w = 0; w < B_; ++w) g += gl[w];
        g *= (1.0f / B_);
        const float r = acc[0] * (1.0f / NPAIR);
        out[0] = 0.7f * r + 0.3f * g;   // total_loss
        out[1] = r;                     // residue_loss
        out[2] = g;                     // global_loss
    }
}

extern "C" void kernel_launch(void* const* d_in, const int* in_sizes, int n_in,
                              void* d_out, int out_size, void* d_ws, size_t ws_size,
                              hipStream_t stream) {
    const float* e1    = (const float*)d_in[0];  // new_emb1   [8,2048,128] f32
    const float* e2    = (const float*)d_in[1];  // new_emb2   [8,2048,128] f32
    const float* g1    = (const float*)d_in[2];  // global_emb1 [8,128]
    const float* g2    = (const float*)d_in[3];  // global_emb2 [8,128]
    const float* lddt  = (const float*)d_in[4];  // lddt_scores [8,2048]
    // d_in[5] = alignment_mask (unused by reference math)
    const int*   align = (const int*)d_in[6];    // alignment   [8,1024,2] i32

    float* acc = (float*)d_ws;
    float* out = (float*)d_out;

    sll_zero_ws<<<1, 1, 0, stream>>>(acc);

    const int tiles  = NPAIR / TILE;          // 512 wave-tiles
    const int blocks = tiles / WPB;           // 256 blocks of 64 threads
    sll_residue_kernel<<<blocks, WPB * 32, 0, stream>>>(e1, e2, lddt, align, acc);

    sll_finalize_kernel<<<1, 256, 0, stream>>>(g1, g2, acc, out);
}